// LIF_24309514895924
// MI455X (gfx1250) — compile-verified
//
#include <hip/hip_runtime.h>
#include <stdint.h>

// ---------------------------------------------------------------------------
// LIF forward scan, MI455X (gfx1250, wave32).
//
// Roofline: ~537 MB of streamed traffic, ~0.27 GFLOP -> 0.5 FLOP/byte, floor
// ~23 us at 23.3 TB/s. Pure bandwidth problem; the serial recurrence over T
// is nonlinear (threshold + reset) so it cannot be blocked into matrix ops.
// CDNA5 features used: GLOBAL_LOAD_ASYNC_TO_LDS_B128 (ASYNCcnt DMA pipeline,
// 8 deep, per-lane private slots -> no barriers, only s_wait_asynccnt),
// ds_load_b128 readback, non-temporal b128 stores (don't thrash 192MB L2).
//
// Main loop is unrolled by DEPTH so LDS slot offsets are immediates and the
// prefetch/store addresses are pure pointer increments; a short clamped
// remainder loop handles the pipeline tail with a constant wait immediate
// (dummy re-issues of the last row keep ASYNCcnt math uniform).
// ---------------------------------------------------------------------------

typedef float f32x4 __attribute__((ext_vector_type(4)));

#define LIF_BLOCK 256
#define LIF_DEPTH 8  // outstanding async loads per lane (ASYNCcnt is 6 bits)

template <int N>
__device__ __forceinline__ void lif_wait_asynccnt() {
#if __has_builtin(__builtin_amdgcn_s_wait_asynccnt)
    __builtin_amdgcn_s_wait_asynccnt(N);
#else
    asm volatile("s_wait_asynccnt %0" ::"n"(N) : "memory");
#endif
}

// GLOBAL_LOAD_ASYNC_TO_LDS_B128 (GV mode): vdst = per-lane LDS byte offset,
// vaddr = 64-bit global address, saddr = off. Tracked with ASYNCcnt.
__device__ __forceinline__ void lif_async_load_b128(uint32_t lds_off,
                                                    const void* gaddr) {
    asm volatile("global_load_async_to_lds_b128 %0, %1, off"
                 :
                 : "v"(lds_off), "v"(gaddr)
                 : "memory");
}

__device__ __forceinline__ f32x4 lif_step(f32x4& v, f32x4 xv, float decay,
                                          float thr) {
    v = decay * v + xv;  // leaky integrate
    f32x4 s;
    s.x = (v.x >= thr) ? 1.0f : 0.0f;  // v - thr >= 0  <=>  v >= thr
    s.y = (v.y >= thr) ? 1.0f : 0.0f;
    s.z = (v.z >= thr) ? 1.0f : 0.0f;
    s.w = (v.w >= thr) ? 1.0f : 0.0f;
    v = v - s * thr;  // soft reset
    return s;
}

__global__ __launch_bounds__(LIF_BLOCK) void lif_fwd_kernel(
    const float* __restrict__ x,         // [T, BN]
    const float* __restrict__ v0,        // [BN]
    const float* __restrict__ p_decay,   // [1]
    const float* __restrict__ p_thresh,  // [1]
    float* __restrict__ out,             // [T, BN]
    int T, int BN) {
    // stage[slot][lane]: per-lane private 16B staging slots.
    __shared__ f32x4 stage[LIF_DEPTH][LIF_BLOCK];

    const int tid = threadIdx.x;
    const int ncol4 = BN >> 2;
    int col4 = blockIdx.x * LIF_BLOCK + tid;
    // Clamp instead of early-exit: async ops honor EXEC, keep all lanes live.
    if (col4 >= ncol4) col4 = ncol4 - 1;

    const float decay = p_decay[0];
    const float thr = p_thresh[0];

    f32x4 v = reinterpret_cast<const f32x4*>(v0)[col4];

    const float* gcol = x + 4 * (size_t)col4;  // this lane's column, t = 0

    // LDS byte offset of this lane's slot 0 (low 32 bits of the generic
    // address of a __shared__ object = wave-relative LDS offset).
    const uint32_t lds0 = (uint32_t)(uintptr_t)&stage[0][tid];
    const uint32_t slot_stride = LIF_BLOCK * (uint32_t)sizeof(f32x4);

    const int Tm1 = T - 1;

    // ---- Prologue: issue rows 0 .. DEPTH-2 into slots 0 .. DEPTH-2. ----
#pragma unroll
    for (int p = 0; p < LIF_DEPTH - 1; ++p) {
        const int tp = (p < Tm1) ? p : Tm1;
        lif_async_load_b128(lds0 + (uint32_t)p * slot_stride,
                            gcol + (size_t)tp * BN);
    }

    int t = 0;
    // Incrementally maintained pointers for the fast path.
    const float* gpre = gcol + (size_t)(LIF_DEPTH - 1) * BN;  // row t+7
    float* gout = out + 4 * (size_t)col4;                     // row t

    // ---- Main loop: unrolled by DEPTH; all prefetches in range while
    //      t + 2*DEPTH - 1 <= T - 1. Slot indices are immediates. ----
    for (; t + 2 * LIF_DEPTH - 1 < T; t += LIF_DEPTH) {
#pragma unroll
        for (int u = 0; u < LIF_DEPTH; ++u) {
            // Issue row t+u+DEPTH-1 into slot (u+DEPTH-1)&(DEPTH-1).
            lif_async_load_b128(
                lds0 + (uint32_t)(((u + LIF_DEPTH - 1) & (LIF_DEPTH - 1)) *
                                  slot_stride),
                gpre);
            gpre += BN;
            // In-order completion: <= DEPTH-1 outstanding => slot u landed.
            lif_wait_asynccnt<LIF_DEPTH - 1>();

            const f32x4 xv = stage[u][tid];  // ds_load_b128
            const f32x4 s = lif_step(v, xv, decay, thr);

            __builtin_nontemporal_store(s, reinterpret_cast<f32x4*>(gout));
            gout += BN;
        }
    }

    // ---- Remainder (< 2*DEPTH iterations): clamped prefetch keeps the
    //      ASYNCcnt arithmetic (and the wait immediate) uniform; dummy
    //      copies land in slots that are never read again. ----
    for (; t < T; ++t) {
        const int tpre = t + (LIF_DEPTH - 1);
        const int tp = (tpre < Tm1) ? tpre : Tm1;
        lif_async_load_b128(
            lds0 + (uint32_t)((tpre & (LIF_DEPTH - 1)) * slot_stride),
            gcol + (size_t)tp * BN);
        lif_wait_asynccnt<LIF_DEPTH - 1>();

        const f32x4 xv = stage[t & (LIF_DEPTH - 1)][tid];
        const f32x4 s = lif_step(v, xv, decay, thr);

        __builtin_nontemporal_store(
            s, reinterpret_cast<f32x4*>(out + (size_t)t * BN) + col4);
    }
}

extern "C" void kernel_launch(void* const* d_in, const int* in_sizes, int n_in,
                              void* d_out, int out_size, void* d_ws,
                              size_t ws_size, hipStream_t stream) {
    const float* x = (const float*)d_in[0];      // x_seq [T,B,N] f32
    const float* v0 = (const float*)d_in[1];     // v0    [B*N]   f32
    const float* decay = (const float*)d_in[2];  // scalar
    const float* thr = (const float*)d_in[3];    // scalar
    // d_in[4] = alpha: only used by the backward pass, unused here.
    float* out = (float*)d_out;

    const int BN = in_sizes[1];
    const int T = in_sizes[0] / BN;
    const int ncol4 = BN / 4;
    const int blocks = (ncol4 + LIF_BLOCK - 1) / LIF_BLOCK;

    lif_fwd_kernel<<<blocks, LIF_BLOCK, 0, stream>>>(x, v0, decay, thr, out, T,
                                                     BN);
}